// GCN_47004122087951
// MI455X (gfx1250) — compile-verified
//
#include <hip/hip_runtime.h>
#include <hip/hip_bf16.h>
#include <math.h>

// ---------------------------------------------------------------------------
// GraphSAGE forward on MI455X (gfx1250, wave32).
// GEMMs use V_WMMA_F32_16X16X4_F32 (fp32 tensor path): workload is L2/atomic
// bound (28.6 GFLOP total), so fp32 WMMA gives full precision at zero cost.
// ---------------------------------------------------------------------------

typedef __attribute__((ext_vector_type(2))) float v2f;
typedef __attribute__((ext_vector_type(8))) float v8f;

#define NN    100000
#define EE    1600000
#define INC   256
#define HIDC  128
#define OUTC  47

// ---------------- A / B fragment loaders (ISA 7.12.2 layouts) --------------
// A 16x4 f32: lanes 0-15 hold {K=k,k+1}, lanes 16-31 hold {K=k+2,k+3}, M=lane&15
__device__ __forceinline__ v2f load_a_frag(const float* __restrict__ A, int lda,
                                           int r0, int k, int lane) {
  const int m  = r0 + (lane & 15);
  const int kk = k + ((lane >> 4) << 1);
  const float* p = A + (size_t)m * lda + kk;
  v2f a; a.x = p[0]; a.y = p[1];
  return a;
}

// B 4x16 f32 (row = K striped across VGPRs, N across lanes), masked on ncols
__device__ __forceinline__ v2f load_b_frag(const float* __restrict__ B, int ldb,
                                           int k, int c0, int lane, int ncols) {
  const int n  = c0 + (lane & 15);
  const int kk = k + ((lane >> 4) << 1);
  v2f b;
  if (n < ncols) {
    b.x = B[(size_t)kk * ldb + n];
    b.y = B[(size_t)(kk + 1) * ldb + n];
  } else {
    b.x = 0.0f; b.y = 0.0f;
  }
  return b;
}

#define WMMA_F32(a, b, c) \
  __builtin_amdgcn_wmma_f32_16x16x4_f32(false, (a), false, (b), (short)0, (c), false, false)

// ------------------------------ utility ------------------------------------
__global__ void zero_kernel(float* __restrict__ p, int n) {
  int i = blockIdx.x * blockDim.x + threadIdx.x;
  if (i < n) p[i] = 0.0f;
}

__global__ void deg_accum_kernel(const long long* __restrict__ dst,
                                 int* __restrict__ deg, int nedges) {
  int e = blockIdx.x * blockDim.x + threadIdx.x;
  if (e < nedges)
    __hip_atomic_fetch_add(&deg[(int)dst[e]], 1, __ATOMIC_RELAXED,
                           __HIP_MEMORY_SCOPE_AGENT);
}

__global__ void deg_finalize_kernel(const int* __restrict__ deg,
                                    float* __restrict__ deg_inv, int n) {
  int i = blockIdx.x * blockDim.x + threadIdx.x;
  if (i < n) {
    int d = deg[i];
    deg_inv[i] = 1.0f / (float)(d > 1 ? d : 1);
  }
}

// --------------------- mean aggregation (wave per edge) --------------------
// agg[dst] += h[src] * deg_inv[dst]; h/agg are L2-resident (51 MB each).
__global__ void aggregate_kernel(const long long* __restrict__ src,
                                 const long long* __restrict__ dst,
                                 const float* __restrict__ h,
                                 const float* __restrict__ deg_inv,
                                 float* __restrict__ agg, int nedges) {
  const int lane = threadIdx.x & 31;
  const int e = (blockIdx.x * blockDim.x + threadIdx.x) >> 5;
  if (e >= nedges) return;
  const int s = (int)src[e];
  const int d = (int)dst[e];
  const float dinv = deg_inv[d];
  const float4 v = ((const float4*)(h + (size_t)s * HIDC))[lane];
  float* ap = agg + (size_t)d * HIDC + lane * 4;
  __hip_atomic_fetch_add(ap + 0, v.x * dinv, __ATOMIC_RELAXED, __HIP_MEMORY_SCOPE_AGENT);
  __hip_atomic_fetch_add(ap + 1, v.y * dinv, __ATOMIC_RELAXED, __HIP_MEMORY_SCOPE_AGENT);
  __hip_atomic_fetch_add(ap + 2, v.z * dinv, __ATOMIC_RELAXED, __HIP_MEMORY_SCOPE_AGENT);
  __hip_atomic_fetch_add(ap + 3, v.w * dinv, __ATOMIC_RELAXED, __HIP_MEMORY_SCOPE_AGENT);
}

// -------- input projection: inp = x@W_in + b_in ; h = relu(inp) ------------
// 128 threads = 4 waves/block; each wave owns a 16-row strip x 128 cols.
__global__ __launch_bounds__(128)
void gemm_in_kernel(const float* __restrict__ x,
                    const float* __restrict__ W,      // 256 x 128
                    const float* __restrict__ bias,   // 128
                    float* __restrict__ inp,
                    float* __restrict__ h, int nrows) {
  const int lane = threadIdx.x & 31;
  const int wave = threadIdx.x >> 5;
  const int r0 = (blockIdx.x * 4 + wave) * 16;
  if (r0 >= nrows) return;   // wave-uniform: EXEC stays all-ones for WMMA

  v8f acc[8] = {};
#pragma unroll 4
  for (int k = 0; k < INC; k += 4) {
    v2f a = load_a_frag(x, INC, r0, k, lane);
#pragma unroll
    for (int t = 0; t < 8; ++t) {
      v2f b = load_b_frag(W, HIDC, k, t * 16, lane, HIDC);
      acc[t] = WMMA_F32(a, b, acc[t]);
    }
  }

  const int rbase = r0 + ((lane >> 4) << 3);
#pragma unroll
  for (int t = 0; t < 8; ++t) {
    const int c = t * 16 + (lane & 15);
    const float bc = bias[c];
#pragma unroll
    for (int v = 0; v < 8; ++v) {
      const size_t idx = (size_t)(rbase + v) * HIDC + c;
      const float val = acc[t][v] + bc;
      inp[idx] = val;
      h[idx] = fmaxf(val, 0.0f);
    }
  }
}

// ----- hidden SAGE conv: out = relu(agg@Wl + h@Wr + bl) + 0.2*inp ----------
__global__ __launch_bounds__(128)
void sage_hidden_kernel(const float* __restrict__ agg,
                        const float* __restrict__ h,
                        const float* __restrict__ Wl,   // 128 x 128
                        const float* __restrict__ Wr,   // 128 x 128
                        const float* __restrict__ bias, // 128
                        const float* __restrict__ resid,
                        float* __restrict__ out, int nrows) {
  const int lane = threadIdx.x & 31;
  const int wave = threadIdx.x >> 5;
  const int r0 = (blockIdx.x * 4 + wave) * 16;
  if (r0 >= nrows) return;

  v8f acc[8] = {};
#pragma unroll 2
  for (int k = 0; k < HIDC; k += 4) {
    v2f a1 = load_a_frag(agg, HIDC, r0, k, lane);
    v2f a2 = load_a_frag(h,   HIDC, r0, k, lane);
#pragma unroll
    for (int t = 0; t < 8; ++t) {
      v2f b1 = load_b_frag(Wl, HIDC, k, t * 16, lane, HIDC);
      acc[t] = WMMA_F32(a1, b1, acc[t]);
      v2f b2 = load_b_frag(Wr, HIDC, k, t * 16, lane, HIDC);
      acc[t] = WMMA_F32(a2, b2, acc[t]);
    }
  }

  const int rbase = r0 + ((lane >> 4) << 3);
#pragma unroll
  for (int t = 0; t < 8; ++t) {
    const int c = t * 16 + (lane & 15);
    const float bc = bias[c];
#pragma unroll
    for (int v = 0; v < 8; ++v) {
      const size_t idx = (size_t)(rbase + v) * HIDC + c;
      float val = fmaxf(acc[t][v] + bc, 0.0f);
      out[idx] = val + 0.2f * resid[idx];
    }
  }
}

// --------- output conv: logits = agg@Wl_o + bl_o + h@Wr_o (N x 47) ---------
__global__ __launch_bounds__(128)
void sage_out_kernel(const float* __restrict__ agg,
                     const float* __restrict__ h,
                     const float* __restrict__ Wl,   // 128 x 47
                     const float* __restrict__ Wr,   // 128 x 47
                     const float* __restrict__ bias, // 47
                     float* __restrict__ out, int nrows) {
  const int lane = threadIdx.x & 31;
  const int wave = threadIdx.x >> 5;
  const int r0 = (blockIdx.x * 4 + wave) * 16;
  if (r0 >= nrows) return;

  v8f acc[3] = {};   // 48 padded cols -> 3 tiles
#pragma unroll 2
  for (int k = 0; k < HIDC; k += 4) {
    v2f a1 = load_a_frag(agg, HIDC, r0, k, lane);
    v2f a2 = load_a_frag(h,   HIDC, r0, k, lane);
#pragma unroll
    for (int t = 0; t < 3; ++t) {
      v2f b1 = load_b_frag(Wl, OUTC, k, t * 16, lane, OUTC);
      acc[t] = WMMA_F32(a1, b1, acc[t]);
      v2f b2 = load_b_frag(Wr, OUTC, k, t * 16, lane, OUTC);
      acc[t] = WMMA_F32(a2, b2, acc[t]);
    }
  }

  const int rbase = r0 + ((lane >> 4) << 3);
#pragma unroll
  for (int t = 0; t < 3; ++t) {
    const int c = t * 16 + (lane & 15);
    if (c < OUTC) {
      const float bc = bias[c];
#pragma unroll
      for (int v = 0; v < 8; ++v)
        out[(size_t)(rbase + v) * OUTC + c] = acc[t][v] + bc;
    }
  }
}

// ------------------- in-place log_softmax (wave per row) -------------------
__global__ void log_softmax_kernel(float* __restrict__ out, int nrows) {
  const int lane = threadIdx.x & 31;
  const int row = (blockIdx.x * blockDim.x + threadIdx.x) >> 5;
  if (row >= nrows) return;
  float* p = out + (size_t)row * OUTC;
  const float v0 = (lane < OUTC) ? p[lane] : -INFINITY;
  const float v1 = (lane + 32 < OUTC) ? p[lane + 32] : -INFINITY;
  float m = fmaxf(v0, v1);
#pragma unroll
  for (int off = 16; off; off >>= 1) m = fmaxf(m, __shfl_xor(m, off, 32));
  float s = ((lane < OUTC) ? __expf(v0 - m) : 0.0f) +
            ((lane + 32 < OUTC) ? __expf(v1 - m) : 0.0f);
#pragma unroll
  for (int off = 16; off; off >>= 1) s += __shfl_xor(s, off, 32);
  const float lse = m + __logf(s);
  if (lane < OUTC) p[lane] = v0 - lse;
  if (lane + 32 < OUTC) p[lane + 32] = v1 - lse;
}

// ------------------------------- launcher ----------------------------------
extern "C" void kernel_launch(void* const* d_in, const int* in_sizes, int n_in,
                              void* d_out, int out_size, void* d_ws, size_t ws_size,
                              hipStream_t stream) {
  (void)in_sizes; (void)n_in; (void)out_size; (void)ws_size;

  const float*     x    = (const float*)d_in[0];
  const long long* ei   = (const long long*)d_in[1];   // [2, E] int64
  const float*     W_in = (const float*)d_in[2];
  const float*     b_in = (const float*)d_in[3];
  const float*     Wl_h = (const float*)d_in[4];       // [3,128,128]
  const float*     bl_h = (const float*)d_in[5];       // [3,128]
  const float*     Wr_h = (const float*)d_in[6];       // [3,128,128]
  const float*     Wl_o = (const float*)d_in[7];       // [128,47]
  const float*     bl_o = (const float*)d_in[8];       // [47]
  const float*     Wr_o = (const float*)d_in[9];       // [128,47]
  float*           out  = (float*)d_out;

  const long long* esrc = ei;
  const long long* edst = ei + EE;

  // Workspace layout (floats): deg(N as int) | deg_inv(N) | inp | hA | hB | agg
  float* ws      = (float*)d_ws;
  int*   deg     = (int*)ws;
  float* deg_inv = ws + NN;
  float* inp     = ws + 2 * (size_t)NN;
  float* hA      = inp + (size_t)NN * HIDC;
  float* hB      = hA  + (size_t)NN * HIDC;
  float* agg     = hB  + (size_t)NN * HIDC;

  const int hElems = NN * HIDC;                         // 12.8M
  const int zb256  = 256;

  // 1) degrees
  zero_kernel<<<(NN + zb256 - 1) / zb256, zb256, 0, stream>>>((float*)deg, NN);
  deg_accum_kernel<<<(EE + zb256 - 1) / zb256, zb256, 0, stream>>>(edst, deg, EE);
  deg_finalize_kernel<<<(NN + zb256 - 1) / zb256, zb256, 0, stream>>>(deg, deg_inv, NN);

  // 2) input projection (residual + relu)
  const int gemmBlocks = (NN + 63) / 64;                // 4 waves x 16 rows per block
  gemm_in_kernel<<<gemmBlocks, 128, 0, stream>>>(x, W_in, b_in, inp, hA, NN);

  // 3) hidden SAGE convs (ping-pong hA/hB)
  float* hcur = hA;
  float* hnxt = hB;
  const int aggBlocks = (hElems + zb256 - 1) / zb256;
  const int edgeBlocks = (EE + 7) / 8;                  // 8 waves/block, wave per edge
  for (int layer = 0; layer < 3; ++layer) {
    zero_kernel<<<aggBlocks, zb256, 0, stream>>>(agg, hElems);
    aggregate_kernel<<<edgeBlocks, 256, 0, stream>>>(esrc, edst, hcur, deg_inv, agg, EE);
    sage_hidden_kernel<<<gemmBlocks, 128, 0, stream>>>(
        agg, hcur,
        Wl_h + (size_t)layer * HIDC * HIDC,
        Wr_h + (size_t)layer * HIDC * HIDC,
        bl_h + (size_t)layer * HIDC,
        inp, hnxt, NN);
    float* t = hcur; hcur = hnxt; hnxt = t;
  }

  // 4) output conv + log_softmax (in place in d_out)
  zero_kernel<<<aggBlocks, zb256, 0, stream>>>(agg, hElems);
  aggregate_kernel<<<edgeBlocks, 256, 0, stream>>>(esrc, edst, hcur, deg_inv, agg, EE);
  sage_out_kernel<<<gemmBlocks, 128, 0, stream>>>(agg, hcur, Wl_o, Wr_o, bl_o, out, NN);
  log_softmax_kernel<<<(NN + 7) / 8, 256, 0, stream>>>(out, NN);
}